// KernelGenerator_66271345377501
// MI455X (gfx1250) — compile-verified
//
#include <hip/hip_runtime.h>
#include <math.h>

typedef __attribute__((ext_vector_type(2))) float v2f;
typedef __attribute__((ext_vector_type(8))) float v8f;

#define G_   10
#define HID  64
#define FFN  128
#define NH   8
#define E_   131072

// packed weight layout (float offsets, per group)
#define OFF_WIN1   0        // 8 Ntile * 16 kstep * 64  = 8192
#define OFF_WOUT1  8192     // 4 Ntile * 32 kstep * 64  = 8192
#define OFF_WIN2   16384
#define OFF_WOUT2  24576
#define OFF_WFC    32768    // 1 Ntile * 16 kstep * 64  = 1024 (cols 8..15 zero)
#define OFF_BIN1   33792    // 128
#define OFF_BOUT1  33920    // 64
#define OFF_BIN2   33984    // 128
#define OFF_BOUT2  34112    // 64
#define OFF_BFC    34176    // 16 (padded)
#define GSTRIDE    34192

#define SCR_STRIDE 18       // 16x16 tile with padded row stride (bank-conflict relief)
#define SCR_SIZE   (16*SCR_STRIDE)

// ---------------- weight pre-pack kernel -------------------------------------
// Packs each 16x16xK B operand into per-fragment lane order so the main kernel
// loads a whole B fragment with one coalesced global_load_b64 per lane.
// frag layout: packed[((tile*KSTEPS + kk)*32 + lane)*2 + v] = W[4kk+2*(lane>>4)+v][16*tile + (lane&15)]
__global__ void pack_weights(const float* __restrict__ w_in1, const float* __restrict__ b_in1,
                             const float* __restrict__ w_out1, const float* __restrict__ b_out1,
                             const float* __restrict__ w_in2, const float* __restrict__ b_in2,
                             const float* __restrict__ w_out2, const float* __restrict__ b_out2,
                             const float* __restrict__ w_fc, const float* __restrict__ b_fc,
                             float* __restrict__ ws) {
  int tid = blockIdx.x * blockDim.x + threadIdx.x;
  if (tid >= G_ * GSTRIDE) return;
  int g = tid / GSTRIDE;
  int pos = tid - g * GSTRIDE;
  float val = 0.0f;
  if (pos < OFF_WOUT1) {                       // w_in1: [64 x 128]
    int u = pos - OFF_WIN1;
    int v = u & 1, L = (u >> 1) & 31, r = u >> 6;
    int kk = r & 15, t = r >> 4;
    int k = 4 * kk + 2 * (L >> 4) + v;
    int n = 16 * t + (L & 15);
    val = w_in1[(g * HID + k) * FFN + n];
  } else if (pos < OFF_WIN2) {                 // w_out1: [128 x 64]
    int u = pos - OFF_WOUT1;
    int v = u & 1, L = (u >> 1) & 31, r = u >> 6;
    int kk = r & 31, t = r >> 5;
    int k = 4 * kk + 2 * (L >> 4) + v;
    int n = 16 * t + (L & 15);
    val = w_out1[(g * FFN + k) * HID + n];
  } else if (pos < OFF_WOUT2) {                // w_in2
    int u = pos - OFF_WIN2;
    int v = u & 1, L = (u >> 1) & 31, r = u >> 6;
    int kk = r & 15, t = r >> 4;
    int k = 4 * kk + 2 * (L >> 4) + v;
    int n = 16 * t + (L & 15);
    val = w_in2[(g * HID + k) * FFN + n];
  } else if (pos < OFF_WFC) {                  // w_out2
    int u = pos - OFF_WOUT2;
    int v = u & 1, L = (u >> 1) & 31, r = u >> 6;
    int kk = r & 31, t = r >> 5;
    int k = 4 * kk + 2 * (L >> 4) + v;
    int n = 16 * t + (L & 15);
    val = w_out2[(g * FFN + k) * HID + n];
  } else if (pos < OFF_BIN1) {                 // w_fc: [64 x 8] padded to 16 cols
    int u = pos - OFF_WFC;
    int v = u & 1, L = (u >> 1) & 31, kk = u >> 6;
    int k = 4 * kk + 2 * (L >> 4) + v;
    int c = L & 15;
    val = (c < NH) ? w_fc[(g * HID + k) * NH + c] : 0.0f;
  } else if (pos < OFF_BOUT1) {
    val = b_in1[g * FFN + (pos - OFF_BIN1)];
  } else if (pos < OFF_BIN2) {
    val = b_out1[g * HID + (pos - OFF_BOUT1)];
  } else if (pos < OFF_BOUT2) {
    val = b_in2[g * FFN + (pos - OFF_BIN2)];
  } else if (pos < OFF_BFC) {
    val = b_out2[g * HID + (pos - OFF_BOUT2)];
  } else {
    int j = pos - OFF_BFC;
    val = (j < NH) ? b_fc[g * NH + j] : 0.0f;
  }
  ws[tid] = val;
}

// ---------------- main fused kernel ------------------------------------------
__device__ __forceinline__ v8f wmma4(v2f a, v2f b, v8f c) {
  // D = A(16x4,f32) * B(4x16,f32) + C(16x16,f32)
  return __builtin_amdgcn_wmma_f32_16x16x4_f32(false, a, false, b, (short)0, c,
                                               false, false);
}

// One MLP block: x = x + relu(x @ w_in + b_in) @ w_out + b_out
// xa: A-fragments of x for two M=16 edge tiles (16 k-steps each, K=HID=64)
__device__ __forceinline__ void mlp_block(v2f (&xa)[2][16],
                                          const float* __restrict__ win,
                                          const float* __restrict__ wout,
                                          const float* __restrict__ bin,
                                          const float* __restrict__ bout,
                                          float* scr0, float* scr1,
                                          int lane, int c, int m, int hi) {
  v8f acc2[2][4];
  #pragma unroll
  for (int i = 0; i < 2; ++i)
    #pragma unroll
    for (int nt = 0; nt < 4; ++nt) acc2[i][nt] = (v8f){};

  for (int t = 0; t < 8; ++t) {                  // FFN N-tiles
    v8f h0 = {}, h1 = {};
    #pragma unroll
    for (int kk = 0; kk < 16; ++kk) {            // K = 64 over x
      v2f bf = *(const v2f*)(win + ((t * 16 + kk) * 32 + lane) * 2);
      h0 = wmma4(xa[0][kk], bf, h0);
      h1 = wmma4(xa[1][kk], bf, h1);
    }
    float bb = bin[16 * t + c];                  // bias depends only on N=c
    #pragma unroll
    for (int j = 0; j < 8; ++j) {
      h0[j] = fmaxf(h0[j] + bb, 0.0f);
      h1[j] = fmaxf(h1[j] + bb, 0.0f);
    }
    // C-layout -> A-layout transpose through per-wave LDS scratch
    #pragma unroll
    for (int j = 0; j < 8; ++j) {
      scr0[(j + 8 * hi) * SCR_STRIDE + c] = h0[j];
      scr1[(j + 8 * hi) * SCR_STRIDE + c] = h1[j];
    }
    __asm__ volatile("" ::: "memory");           // DS in-order per wave
    v2f ha0[4], ha1[4];
    #pragma unroll
    for (int q = 0; q < 4; ++q) {
      ha0[q] = *(const v2f*)(scr0 + m * SCR_STRIDE + 4 * q + 2 * hi);
      ha1[q] = *(const v2f*)(scr1 + m * SCR_STRIDE + 4 * q + 2 * hi);
    }
    __asm__ volatile("" ::: "memory");
    #pragma unroll
    for (int q = 0; q < 4; ++q) {                // K-steps t*4+q of GEMM2
      #pragma unroll
      for (int nt = 0; nt < 4; ++nt) {           // HID N-tiles
        v2f bf = *(const v2f*)(wout + ((nt * 32 + t * 4 + q) * 32 + lane) * 2);
        acc2[0][nt] = wmma4(ha0[q], bf, acc2[0][nt]);
        acc2[1][nt] = wmma4(ha1[q], bf, acc2[1][nt]);
      }
    }
  }
  // bias (C layout) then transpose + residual (A layout), updating xa in place
  #pragma unroll
  for (int nt = 0; nt < 4; ++nt) {
    float bb = bout[16 * nt + c];
    #pragma unroll
    for (int j = 0; j < 8; ++j) {
      scr0[(j + 8 * hi) * SCR_STRIDE + c] = acc2[0][nt][j] + bb;
      scr1[(j + 8 * hi) * SCR_STRIDE + c] = acc2[1][nt][j] + bb;
    }
    __asm__ volatile("" ::: "memory");
    #pragma unroll
    for (int q = 0; q < 4; ++q) {
      v2f n0 = *(const v2f*)(scr0 + m * SCR_STRIDE + 4 * q + 2 * hi);
      v2f n1 = *(const v2f*)(scr1 + m * SCR_STRIDE + 4 * q + 2 * hi);
      xa[0][4 * nt + q] += n0;
      xa[1][4 * nt + q] += n1;
    }
    __asm__ volatile("" ::: "memory");
  }
}

__global__ __launch_bounds__(256) void fused_edge_mlp(
    const float* __restrict__ edge_attr, const int* __restrict__ edge_index,
    const float* __restrict__ deg, const float* __restrict__ deg_coef,
    const float* __restrict__ ws, float* __restrict__ out) {
  __shared__ float scr[8][2][SCR_SIZE];
  const int w = threadIdx.x >> 5;
  const int lane = threadIdx.x & 31;
  const int c = lane & 15;      // N column / M row within half-wave
  const int hi = lane >> 4;     // lane half select (K split)
  const int m = c;

  const int task = blockIdx.x;              // 0 .. G*512-1
  const int g = task >> 9;
  const int chunk = task & 511;
  const int e0 = chunk * 256 + w * 32;      // 32 edges per wave

  const float* wb = ws + g * GSTRIDE;
  float* scr0 = scr[w][0];
  float* scr1 = scr[w][1];

  // ---- load x A-fragments with degree scaling:
  // x = attr * (dc0 + sqrt(deg[dst]) * dc1)
  const float* dc0 = deg_coef + g * HID;
  const float* dc1 = deg_coef + (G_ + g) * HID;
  const int em0 = e0 + m;
  const int em1 = e0 + 16 + m;
  const float sd0 = sqrtf(deg[edge_index[E_ + em0]]);
  const float sd1 = sqrtf(deg[edge_index[E_ + em1]]);

  v2f xa[2][16];
  #pragma unroll
  for (int kk = 0; kk < 16; ++kk) {
    int k = 4 * kk + 2 * hi;
    v2f d0 = *(const v2f*)(dc0 + k);
    v2f d1 = *(const v2f*)(dc1 + k);
    v2f a0 = *(const v2f*)(edge_attr + (size_t)em0 * HID + k);
    v2f a1 = *(const v2f*)(edge_attr + (size_t)em1 * HID + k);
    xa[0][kk] = a0 * (d0 + sd0 * d1);
    xa[1][kk] = a1 * (d0 + sd1 * d1);
  }

  // ---- two residual MLP blocks
  mlp_block(xa, wb + OFF_WIN1, wb + OFF_WOUT1, wb + OFF_BIN1, wb + OFF_BOUT1,
            scr0, scr1, lane, c, m, hi);
  mlp_block(xa, wb + OFF_WIN2, wb + OFF_WOUT2, wb + OFF_BIN2, wb + OFF_BOUT2,
            scr0, scr1, lane, c, m, hi);

  // ---- final relu (elementwise, A layout is fine)
  #pragma unroll
  for (int kk = 0; kk < 16; ++kk) {
    #pragma unroll
    for (int v = 0; v < 2; ++v) {
      xa[0][kk][v] = fmaxf(xa[0][kk][v], 0.0f);
      xa[1][kk][v] = fmaxf(xa[1][kk][v], 0.0f);
    }
  }

  // ---- fc projection: x @ w_fc (padded 64x16, cols 8..15 zero)
  v8f f0 = {}, f1 = {};
  #pragma unroll
  for (int kk = 0; kk < 16; ++kk) {
    v2f bf = *(const v2f*)(wb + OFF_WFC + (kk * 32 + lane) * 2);
    f0 = wmma4(xa[0][kk], bf, f0);
    f1 = wmma4(xa[1][kk], bf, f1);
  }

  if (c < NH) {
    float bb = wb[OFF_BFC + c];
    size_t base = (size_t)g * E_;
    #pragma unroll
    for (int j = 0; j < 8; ++j) {
      int r = j + 8 * hi;
      out[(base + e0 + r) * NH + c] = f0[j] + bb;
      out[(base + e0 + 16 + r) * NH + c] = f1[j] + bb;
    }
  }
}

// ---------------- launcher ---------------------------------------------------
extern "C" void kernel_launch(void* const* d_in, const int* in_sizes, int n_in,
                              void* d_out, int out_size, void* d_ws, size_t ws_size,
                              hipStream_t stream) {
  const float* edge_attr  = (const float*)d_in[0];
  const int*   edge_index = (const int*)d_in[1];
  const float* deg        = (const float*)d_in[2];
  const float* deg_coef   = (const float*)d_in[3];
  const float* w_in1  = (const float*)d_in[4];
  const float* b_in1  = (const float*)d_in[5];
  const float* w_out1 = (const float*)d_in[6];
  const float* b_out1 = (const float*)d_in[7];
  const float* w_in2  = (const float*)d_in[8];
  const float* b_in2  = (const float*)d_in[9];
  const float* w_out2 = (const float*)d_in[10];
  const float* b_out2 = (const float*)d_in[11];
  const float* w_fc   = (const float*)d_in[12];
  const float* b_fc   = (const float*)d_in[13];
  float* ws  = (float*)d_ws;
  float* out = (float*)d_out;

  const int packTotal = G_ * GSTRIDE;   // 341,920 floats = 1.37 MB of d_ws
  pack_weights<<<(packTotal + 255) / 256, 256, 0, stream>>>(
      w_in1, b_in1, w_out1, b_out1, w_in2, b_in2, w_out2, b_out2, w_fc, b_fc, ws);

  fused_edge_mlp<<<G_ * (E_ / 256), 256, 0, stream>>>(
      edge_attr, edge_index, deg, deg_coef, ws, out);
}